// QueryAndGroup_37684043055439
// MI455X (gfx1250) — compile-verified
//
#include <hip/hip_runtime.h>

typedef __attribute__((ext_vector_type(2))) float v2f;
typedef __attribute__((ext_vector_type(8))) float v8f;

#define NSAMPLE 32
#define RADIUS2 0.04f   // 0.2^2

__device__ __forceinline__ float lane_bcast_f(float v, int src) {
  return __builtin_bit_cast(float,
      __builtin_amdgcn_ds_bpermute(src << 2, __builtin_bit_cast(int, v)));
}

// ---------------------------------------------------------------------------
// Kernel 1: ball query via V_WMMA_F32_16X16X4_F32.
// One wave32 handles 16 query points; scans N in ascending chunks of 16.
// D[m][n] = A*B + C with A=q (16x4, K=3 padded), B=-2p (4x16),
// C = |p|^2 - (r^2 - |q|^2)  =>  D = d2 - r^2, inside-ball <=> D < 0.
// ---------------------------------------------------------------------------
__global__ __launch_bounds__(256) void ball_query_wmma(
    const float* __restrict__ xyz,      // (B,N,3)
    const float* __restrict__ new_xyz,  // (B,S,3)
    int* __restrict__ idx_out,          // (B,S,NSAMPLE)
    int B, int N, int S)
{
  __shared__ int lds_idx[8 * 16 * NSAMPLE];  // 8 waves * 16 rows * 32 slots
  __shared__ int lds_cnt[8 * 16];

  const int lane = threadIdx.x & 31;
  const int wave = threadIdx.x >> 5;
  const int wid  = blockIdx.x * 8 + wave;
  const int gpw  = S / 16;                   // 16-query groups per batch
  if (wid >= B * gpw) return;                // wave-uniform
  const int b  = wid / gpw;
  const int s0 = (wid % gpw) * 16;

  const int  laneN = lane & 15;
  const bool hi    = lane >= 16;

  int* midx = &lds_idx[wave * 16 * NSAMPLE];
  int* mcnt = &lds_cnt[wave * 16];
  if (lane < 16) mcnt[lane] = 0;

  // Query point for row laneN (replicated across both halves)
  const float* qp = new_xyz + ((size_t)b * S + (s0 + laneN)) * 3;
  const float qx = qp[0], qy = qp[1], qz = qp[2];
  const float r2mq = RADIUS2 - (qx * qx + qy * qy + qz * qz);

  // A matrix 16x4 f32: lanes 0-15 hold K=0,1 ; lanes 16-31 hold K=2,3
  v2f A;
  A.x = hi ? qz : qx;
  A.y = hi ? 0.0f : qy;

  // Per-accumulator-VGPR threshold: row(v) = v + (hi ? 8 : 0)
  float thr[8];
  #pragma unroll
  for (int v = 0; v < 8; ++v)
    thr[v] = lane_bcast_f(r2mq, v + (hi ? 8 : 0));

  const float* xb = xyz + (size_t)b * N * 3;
  const int nchunks = N / 16;

  for (int j = 0; j < nchunks; ++j) {
    const int base = j * 16;
    const float* pp = xb + (size_t)(base + laneN) * 3;
    const float px = pp[0], py = pp[1], pz = pp[2];
    if (j + 1 < nchunks)
      __builtin_prefetch(xb + (size_t)(base + 16 + laneN) * 3, 0, 3);

    const float pn = px * px + py * py + pz * pz;

    // B matrix 4x16 f32 (column laneN), -2 folded in
    v2f Bm;
    Bm.x = hi ? (-2.0f * pz) : (-2.0f * px);
    Bm.y = hi ? 0.0f : (-2.0f * py);

    v8f Cm;
    #pragma unroll
    for (int v = 0; v < 8; ++v) Cm[v] = pn - thr[v];

    // D = d2 - r^2  (one WMMA = 256 pairwise distances)
    v8f D = __builtin_amdgcn_wmma_f32_16x16x4_f32(
        false, A, false, Bm, (short)0, Cm, false, false);

    #pragma unroll
    for (int v = 0; v < 8; ++v) {
      const bool pred = D[v] < 0.0f;
      const unsigned m32 = __builtin_amdgcn_ballot_w32(pred);
      const unsigned mh  = hi ? (m32 >> 16) : (m32 & 0xffffu);
      const int row = v + (hi ? 8 : 0);
      const int cnt = mcnt[row];                       // LDS broadcast read
      if (cnt < NSAMPLE) {                             // uniform per half
        const int pos = cnt + __popc(mh & ((1u << laneN) - 1u));
        if (pred && pos < NSAMPLE)
          midx[row * NSAMPLE + pos] = base + laneN;    // ordered append
        if (laneN == 0) mcnt[row] = cnt + __popc(mh);
      }
    }

    // Uniform early exit: all 16 rows saturated
    const int myc = mcnt[laneN];
    if (__builtin_amdgcn_ballot_w32(myc >= NSAMPLE) == 0xffffffffu) break;
  }

  // Pad with first hit (0 if none) and write indices; lane = sample slot
  #pragma unroll 1
  for (int r = 0; r < 16; ++r) {
    int cnt = mcnt[r];
    cnt = cnt > NSAMPLE ? NSAMPLE : cnt;
    const int first = (cnt > 0) ? midx[r * NSAMPLE] : 0;
    const int vsl   = (lane < cnt) ? midx[r * NSAMPLE + lane] : first;
    idx_out[((size_t)b * S + (s0 + r)) * NSAMPLE + lane] = vsl;
  }
}

// ---------------------------------------------------------------------------
// Kernel 2: gather (xyz - new_xyz) and features into (B, 3+C, S, NSAMPLE).
// One wave per query point; lane = sample k; output stores fully coalesced.
// ---------------------------------------------------------------------------
__global__ __launch_bounds__(256) void group_gather(
    const float* __restrict__ xyz,      // (B,N,3)
    const float* __restrict__ new_xyz,  // (B,S,3)
    const float* __restrict__ feats,    // (B,C,N)
    const int*   __restrict__ idx,      // (B,S,NSAMPLE)
    float* __restrict__ out,            // (B,3+C,S,NSAMPLE)
    int B, int N, int S, int C)
{
  const int lane = threadIdx.x & 31;
  const int wave = threadIdx.x >> 5;
  const int wid  = blockIdx.x * 8 + wave;
  if (wid >= B * S) return;
  const int b = wid / S;
  const int s = wid % S;

  const int k = idx[((size_t)b * S + s) * NSAMPLE + lane];

  const float* pp = xyz + ((size_t)b * N + k) * 3;
  const float* qp = new_xyz + ((size_t)b * S + s) * 3;
  const int CH = C + 3;
  const size_t chstride = (size_t)S * NSAMPLE;
  float* ob = out + ((size_t)b * CH) * chstride + (size_t)s * NSAMPLE + lane;

  ob[0 * chstride] = pp[0] - qp[0];
  ob[1 * chstride] = pp[1] - qp[1];
  ob[2 * chstride] = pp[2] - qp[2];

  const float* fb = feats + (size_t)b * C * N + k;
  float* of = ob + 3 * chstride;
  #pragma unroll 4
  for (int c = 0; c < C; ++c)
    of[(size_t)c * chstride] = fb[(size_t)c * N];
}

// ---------------------------------------------------------------------------
extern "C" void kernel_launch(void* const* d_in, const int* in_sizes, int n_in,
                              void* d_out, int out_size, void* d_ws, size_t ws_size,
                              hipStream_t stream) {
  const float* xyz     = (const float*)d_in[0];
  const float* new_xyz = (const float*)d_in[1];
  const float* feats   = (const float*)d_in[2];

  // Reference shapes: B=8, N=16384, S=1024, C=64
  const int B = 8;
  const int N = in_sizes[0] / (B * 3);
  const int S = in_sizes[1] / (B * 3);
  const int C = in_sizes[2] / (B * N);

  int*   idx = (int*)d_ws;     // B*S*NSAMPLE ints = 1 MB
  float* out = (float*)d_out;

  const int waves1 = B * (S / 16);
  ball_query_wmma<<<dim3((waves1 + 7) / 8), dim3(256), 0, stream>>>(
      xyz, new_xyz, idx, B, N, S);

  const int waves2 = B * S;
  group_gather<<<dim3((waves2 + 7) / 8), dim3(256), 0, stream>>>(
      xyz, new_xyz, feats, idx, out, B, N, S, C);
}